// StandaloneAlignment_4587025072196
// MI455X (gfx1250) — compile-verified
//
#include <hip/hip_runtime.h>
#include <hip/hip_bf16.h>

// ---------------------------------------------------------------------------
// StandaloneAlignment for MI455X (gfx1250, wave32, WMMA).
//
//   q = proj(queries)  [B,128,T1]      k = proj(keys) [B,128,T2]
//   dist = q2 + k2 - 2*q.k ; l = -TEMP*dist
//   logprob = log_softmax(l, T2) + log(prior+1e-8)
//   soft    = softmax(logprob, T2)
//
// All GEMMs use V_WMMA_F32_16X16X4_F32 (full f32 precision on the matrix
// pipe).  All shapes are compile-time constants so every WMMA K-loop fully
// unrolls and every fragment load is a global_load with an immediate offset
// from one folded base pointer (no per-iteration 64-bit address math).
// The qk GEMM + both softmaxes + prior are fused into one kernel so the only
// large HBM traffic is prior-in (67MB) + two outputs (134MB) ~ 9us @ 23.3TB/s.
// ---------------------------------------------------------------------------

typedef __attribute__((ext_vector_type(2))) float v2f;
typedef __attribute__((ext_vector_type(8))) float v8f;

#define TEMP_C      0.0005f
#define EPS_PRIOR_C 1e-8f

constexpr int BB  = 16;    // batch
constexpr int T1C = 2048;  // mel frames
constexpr int T2C = 512;   // text tokens
constexpr int CC  = 128;   // model channels
constexpr int IC  = 384;   // inverted-bottleneck channels

__device__ __forceinline__ v8f wmma_f32_16x16x4(v2f a, v2f b, v8f c) {
  // (neg_a, A, neg_b, B, c_mod, C, reuse_a, reuse_b)
  return __builtin_amdgcn_wmma_f32_16x16x4_f32(false, a, false, b, (short)0, c,
                                               false, false);
}

// out[b,m,t] = sum_k W[m,k] * X[b,k,t] + bias[m]   (O = 128, K = KDIM)
template <int KDIM, int TDIM>
__global__ __launch_bounds__(32) void proj_in_gemm(
    const float* __restrict__ X, const float* __restrict__ W,
    const float* __restrict__ bias, float* __restrict__ out) {
  const int lane = threadIdx.x & 31;
  const int ln = lane & 15, lh = lane >> 4;
  const int n0 = blockIdx.x << 4;
  const int m0 = blockIdx.y << 4;
  const int b  = blockIdx.z;
  // folded bases: a.{x,y} = Wa[k], Wa[k+1] ; bb.{x,y} = Xb[k*T], Xb[(k+1)*T]
  const float* Wa = W + (m0 + ln) * KDIM + (lh << 1);
  const float* Xb = X + (size_t)b * KDIM * TDIM + (size_t)(lh << 1) * TDIM +
                    n0 + ln;
  v8f acc = {};
#pragma unroll
  for (int k = 0; k < KDIM; k += 4) {
    v2f a, bb;
    a.x  = Wa[k];
    a.y  = Wa[k + 1];
    bb.x = Xb[(size_t)k * TDIM];
    bb.y = Xb[(size_t)(k + 1) * TDIM];
    acc = wmma_f32_16x16x4(a, bb, acc);
  }
  float* Ob = out + (size_t)b * CC * TDIM;
#pragma unroll
  for (int v = 0; v < 8; ++v) {
    const int m = m0 + v + (lh << 3);
    Ob[(size_t)m * TDIM + n0 + ln] = acc[v] + bias[m];
  }
}

// Depthwise conv, C=128, kernel 7, pad 3.  x,out: [B,128,T]
__global__ void dwconv7_kernel(const float* __restrict__ x,
                               const float* __restrict__ w,
                               const float* __restrict__ bias,
                               float* __restrict__ out, int T) {
  const int t = blockIdx.x * blockDim.x + threadIdx.x;
  const int c = blockIdx.y;
  const int b = blockIdx.z;
  if (t >= T) return;
  const float* xr = x + ((size_t)b * CC + c) * T;
  float acc = bias[c];
#pragma unroll
  for (int j = 0; j < 7; ++j) {
    const int tt = t + j - 3;
    const float xv = (tt >= 0 && tt < T) ? xr[tt] : 0.0f;
    acc = fmaf(w[c * 7 + j], xv, acc);
  }
  out[((size_t)b * CC + c) * T + t] = acc;
}

// In-place LayerNorm over C (=128) at each (b,t); x layout [B,128,T].
// Each thread owns 4 consecutive t (b128 loads/stores).  block=128 threads.
template <int TDIM>
__global__ void layernorm_c_kernel(float* __restrict__ x,
                                   const float* __restrict__ g,
                                   const float* __restrict__ bt) {
  const int t4 = (blockIdx.x * blockDim.x + threadIdx.x) << 2;
  const int b  = blockIdx.y;
  float* xb = x + (size_t)b * CC * TDIM + t4;
  float4 s  = {0.f, 0.f, 0.f, 0.f};
  float4 s2 = {0.f, 0.f, 0.f, 0.f};
#pragma unroll 4
  for (int c = 0; c < CC; ++c) {
    const float4 v = *(const float4*)(xb + (size_t)c * TDIM);
    s.x += v.x; s.y += v.y; s.z += v.z; s.w += v.w;
    s2.x = fmaf(v.x, v.x, s2.x); s2.y = fmaf(v.y, v.y, s2.y);
    s2.z = fmaf(v.z, v.z, s2.z); s2.w = fmaf(v.w, v.w, s2.w);
  }
  const float inv = 1.0f / (float)CC;
  float4 m = {s.x * inv, s.y * inv, s.z * inv, s.w * inv};
  float4 r;
  r.x = rsqrtf(s2.x * inv - m.x * m.x + 1e-6f);
  r.y = rsqrtf(s2.y * inv - m.y * m.y + 1e-6f);
  r.z = rsqrtf(s2.z * inv - m.z * m.z + 1e-6f);
  r.w = rsqrtf(s2.w * inv - m.w * m.w + 1e-6f);
#pragma unroll 4
  for (int c = 0; c < CC; ++c) {
    float4 v = *(const float4*)(xb + (size_t)c * TDIM);
    const float gc = g[c], bc = bt[c];
    v.x = (v.x - m.x) * r.x * gc + bc;
    v.y = (v.y - m.y) * r.y * gc + bc;
    v.z = (v.z - m.z) * r.z * gc + bc;
    v.w = (v.w - m.w) * r.w * gc + bc;
    *(float4*)(xb + (size_t)c * TDIM) = v;
  }
}

// hidden[b,m,t] = gelu_exact( sum_k W1[m,k]*X[b,k,t] + b1[m] )  (O=384, K=128)
template <int TDIM>
__global__ __launch_bounds__(32) void gemm_w1_gelu_kernel(
    const float* __restrict__ X, const float* __restrict__ W,
    const float* __restrict__ bias, float* __restrict__ out) {
  const int lane = threadIdx.x & 31;
  const int ln = lane & 15, lh = lane >> 4;
  const int n0 = blockIdx.x << 4;
  const int m0 = blockIdx.y << 4;
  const int b  = blockIdx.z;
  const float* Wa = W + (m0 + ln) * CC + (lh << 1);
  const float* Xb = X + (size_t)b * CC * TDIM + (size_t)(lh << 1) * TDIM +
                    n0 + ln;
  v8f acc = {};
#pragma unroll
  for (int k = 0; k < CC; k += 4) {
    v2f a, bb;
    a.x  = Wa[k];
    a.y  = Wa[k + 1];
    bb.x = Xb[(size_t)k * TDIM];
    bb.y = Xb[(size_t)(k + 1) * TDIM];
    acc = wmma_f32_16x16x4(a, bb, acc);
  }
  float* Ob = out + (size_t)b * IC * TDIM;
#pragma unroll
  for (int v = 0; v < 8; ++v) {
    const int m = m0 + v + (lh << 3);
    float h = acc[v] + bias[m];
    h = 0.5f * h * (1.0f + erff(h * 0.70710678118654752f));  // exact GELU
    Ob[(size_t)m * TDIM + n0 + ln] = h;
  }
}

// gx2[b,i] = sum_t H[b,i,t]^2   (GRN statistics; reduce along T)
__global__ void rowsumsq_kernel(const float* __restrict__ H,
                                float* __restrict__ gx2, int T) {
  const int i = blockIdx.x;  // 0..383
  const int b = blockIdx.y;
  const float* r = H + ((size_t)b * IC + i) * T;
  float s = 0.0f;
  for (int t = threadIdx.x; t < T; t += blockDim.x) {
    const float v = r[t];
    s = fmaf(v, v, s);
  }
  __shared__ float red[256];
  red[threadIdx.x] = s;
  __syncthreads();
  for (int off = 128; off > 0; off >>= 1) {
    if (threadIdx.x < off) red[threadIdx.x] += red[threadIdx.x + off];
    __syncthreads();
  }
  if (threadIdx.x == 0) gx2[b * IC + i] = red[0];
}

// scale[b,i] = 1 + gamma[i] * Gx[b,i] / (mean_i Gx[b,i] + 1e-6)
__global__ void grn_scale_kernel(const float* __restrict__ gx2,
                                 const float* __restrict__ gamma,
                                 float* __restrict__ scale) {
  const int b = blockIdx.x;
  const int i = threadIdx.x;  // 384 threads
  const float gx = sqrtf(gx2[b * IC + i]);
  __shared__ float sgx[IC];
  __shared__ float smean;
  sgx[i] = gx;
  __syncthreads();
  if (i == 0) {
    float s = 0.0f;
    for (int j = 0; j < IC; ++j) s += sgx[j];
    smean = s * (1.0f / (float)IC);
  }
  __syncthreads();
  scale[b * IC + i] = 1.0f + gamma[i] * (gx / (smean + 1e-6f));
}

// out[b,m,t] = res[b,m,t] + b2[m] + sum_i W2[m,i]*(H[b,i,t]*scale[b,i]+grn_b[i])
// (GRN applied as B-prologue; O=128, K=384)
template <int TDIM>
__global__ __launch_bounds__(32) void gemm_w2_grn_res_kernel(
    const float* __restrict__ H, const float* __restrict__ W,
    const float* __restrict__ bias, const float* __restrict__ scale,
    const float* __restrict__ gbeta, const float* __restrict__ res,
    float* __restrict__ out) {
  const int lane = threadIdx.x & 31;
  const int ln = lane & 15, lh = lane >> 4;
  const int n0 = blockIdx.x << 4;
  const int m0 = blockIdx.y << 4;
  const int b  = blockIdx.z;
  const float* Wa = W + (m0 + ln) * IC + (lh << 1);
  const float* Hb = H + (size_t)b * IC * TDIM + (size_t)(lh << 1) * TDIM +
                    n0 + ln;
  const float* sb = scale + b * IC + (lh << 1);
  const float* gb = gbeta + (lh << 1);
  v8f acc = {};
#pragma unroll
  for (int k = 0; k < IC; k += 4) {
    v2f a, bb;
    a.x  = Wa[k];
    a.y  = Wa[k + 1];
    bb.x = fmaf(Hb[(size_t)k * TDIM],       sb[k],     gb[k]);
    bb.y = fmaf(Hb[(size_t)(k + 1) * TDIM], sb[k + 1], gb[k + 1]);
    acc = wmma_f32_16x16x4(a, bb, acc);
  }
  const float* Rb = res + (size_t)b * CC * TDIM;
  float* Ob = out + (size_t)b * CC * TDIM;
#pragma unroll
  for (int v = 0; v < 8; ++v) {
    const int m = m0 + v + (lh << 3);
    Ob[(size_t)m * TDIM + n0 + ln] =
        Rb[(size_t)m * TDIM + n0 + ln] + acc[v] + bias[m];
  }
}

// sq[b,t] = sum_c x[b,c,t]^2 ; 4 t per thread (b128 loads), block=128
template <int TDIM>
__global__ void colsumsq_kernel(const float* __restrict__ x,
                                float* __restrict__ out) {
  const int t4 = (blockIdx.x * blockDim.x + threadIdx.x) << 2;
  const int b  = blockIdx.y;
  const float* xb = x + (size_t)b * CC * TDIM + t4;
  float4 s = {0.f, 0.f, 0.f, 0.f};
#pragma unroll 4
  for (int c = 0; c < CC; ++c) {
    const float4 v = *(const float4*)(xb + (size_t)c * TDIM);
    s.x = fmaf(v.x, v.x, s.x); s.y = fmaf(v.y, v.y, s.y);
    s.z = fmaf(v.z, v.z, s.z); s.w = fmaf(v.w, v.w, s.w);
  }
  *(float4*)(out + (size_t)b * TDIM + t4) = s;
}

// Fused: qk WMMA GEMM (K=128) -> logits -> log_softmax(T2) + log(prior+eps)
//        -> logprob out -> softmax(T2) -> soft out.
// Block = 512 threads (16 waves); wave w owns columns [32w, 32w+32) as two
// 16x16 WMMA accumulator tiles (64 fully-unrolled WMMAs); row reductions via
// half-wave xor shuffles + LDS across the 16 waves.
__global__ __launch_bounds__(512) void attn_fused_kernel(
    const float* __restrict__ Q, const float* __restrict__ K,
    const float* __restrict__ q2, const float* __restrict__ k2,
    const float* __restrict__ prior, float* __restrict__ out_soft,
    float* __restrict__ out_logp) {
  const int tid  = threadIdx.x;
  const int lane = tid & 31;
  const int wv   = tid >> 5;          // 0..15
  const int ln = lane & 15, lh = lane >> 4;
  const int m0 = blockIdx.x << 4;     // t1 tile base
  const int b  = blockIdx.y;
  const int n0 = wv << 5;             // 32 columns per wave
  // folded bases, all K offsets become compile-time immediates
  const float* qa = Q + (size_t)b * CC * T1C + (size_t)(lh << 1) * T1C +
                    m0 + ln;
  const float* kb = K + (size_t)b * CC * T2C + (size_t)(lh << 1) * T2C +
                    n0 + ln;

  v8f acc0 = {}, acc1 = {};
#pragma unroll
  for (int k = 0; k < CC; k += 4) {
    v2f a, b0, b1;
    a.x  = qa[(size_t)k * T1C];           // A = q^T tile (MxK = t1 x c)
    a.y  = qa[(size_t)(k + 1) * T1C];
    b0.x = kb[(size_t)k * T2C];
    b0.y = kb[(size_t)(k + 1) * T2C];
    b1.x = kb[(size_t)k * T2C + 16];
    b1.y = kb[(size_t)(k + 1) * T2C + 16];
    acc0 = wmma_f32_16x16x4(a, b0, acc0);
    acc1 = wmma_f32_16x16x4(a, b1, acc1);
  }

  // logits l = -TEMP * (q2 + k2 - 2*qk), overwrite accumulators
  const float k2v0 = k2[b * T2C + n0 + ln];
  const float k2v1 = k2[b * T2C + n0 + 16 + ln];
#pragma unroll
  for (int v = 0; v < 8; ++v) {
    const int m = m0 + v + (lh << 3);
    const float qq = q2[b * T1C + m];
    acc0[v] = -TEMP_C * (qq + k2v0 - 2.0f * acc0[v]);
    acc1[v] = -TEMP_C * (qq + k2v1 - 2.0f * acc1[v]);
  }

  __shared__ float red[16][17];
  __shared__ float rowz[16];

  // ---- pass 1: logZ1 over the 512 logits of each row ----
#pragma unroll
  for (int v = 0; v < 8; ++v) {
    float mv = fmaxf(acc0[v], acc1[v]);
#pragma unroll
    for (int off = 1; off < 16; off <<= 1)
      mv = fmaxf(mv, __shfl_xor(mv, off, 32));
    if (ln == 0) red[v + (lh << 3)][wv] = mv;
  }
  __syncthreads();
  if (tid < 16) {
    float mv = red[tid][0];
    for (int j = 1; j < 16; ++j) mv = fmaxf(mv, red[tid][j]);
    rowz[tid] = mv;
  }
  __syncthreads();
#pragma unroll
  for (int v = 0; v < 8; ++v) {
    const float rm = rowz[v + (lh << 3)];
    float s = __expf(acc0[v] - rm) + __expf(acc1[v] - rm);
#pragma unroll
    for (int off = 1; off < 16; off <<= 1) s += __shfl_xor(s, off, 32);
    if (ln == 0) red[v + (lh << 3)][wv] = s;
  }
  __syncthreads();
  if (tid < 16) {
    float s = 0.0f;
    for (int j = 0; j < 16; ++j) s += red[tid][j];
    rowz[tid] += __logf(s);   // logZ1 = max + log(sum exp)
  }
  __syncthreads();

  // attn = l - logZ1 + log(prior + eps); emit logprob
#pragma unroll
  for (int v = 0; v < 8; ++v) {
    const int m = m0 + v + (lh << 3);
    const float lz = rowz[v + (lh << 3)];
    const size_t base = ((size_t)b * T1C + m) * T2C;
    acc0[v] = acc0[v] - lz + __logf(prior[base + n0 + ln] + EPS_PRIOR_C);
    acc1[v] = acc1[v] - lz + __logf(prior[base + n0 + 16 + ln] + EPS_PRIOR_C);
    out_logp[base + n0 + ln]      = acc0[v];
    out_logp[base + n0 + 16 + ln] = acc1[v];
  }
  __syncthreads();

  // ---- pass 2: logZ2 over attn, soft = exp(attn - logZ2) ----
#pragma unroll
  for (int v = 0; v < 8; ++v) {
    float mv = fmaxf(acc0[v], acc1[v]);
#pragma unroll
    for (int off = 1; off < 16; off <<= 1)
      mv = fmaxf(mv, __shfl_xor(mv, off, 32));
    if (ln == 0) red[v + (lh << 3)][wv] = mv;
  }
  __syncthreads();
  if (tid < 16) {
    float mv = red[tid][0];
    for (int j = 1; j < 16; ++j) mv = fmaxf(mv, red[tid][j]);
    rowz[tid] = mv;
  }
  __syncthreads();
#pragma unroll
  for (int v = 0; v < 8; ++v) {
    const float rm = rowz[v + (lh << 3)];
    float s = __expf(acc0[v] - rm) + __expf(acc1[v] - rm);
#pragma unroll
    for (int off = 1; off < 16; off <<= 1) s += __shfl_xor(s, off, 32);
    if (ln == 0) red[v + (lh << 3)][wv] = s;
  }
  __syncthreads();
  if (tid < 16) {
    float s = 0.0f;
    for (int j = 0; j < 16; ++j) s += red[tid][j];
    rowz[tid] += __logf(s);
  }
  __syncthreads();
#pragma unroll
  for (int v = 0; v < 8; ++v) {
    const int m = m0 + v + (lh << 3);
    const float lz = rowz[v + (lh << 3)];
    const size_t base = ((size_t)b * T1C + m) * T2C;
    out_soft[base + n0 + ln]      = __expf(acc0[v] - lz);
    out_soft[base + n0 + 16 + ln] = __expf(acc1[v] - lz);
  }
}

// ---------------------------------------------------------------------------
// Host orchestration
// ---------------------------------------------------------------------------
template <int TDIM>
static void launch_convnext_block(const float* x, float* tmp, float* H,
                                  float* gx2, float* scl, const float* dw_w,
                                  const float* dw_b, const float* ln_g,
                                  const float* ln_b, const float* w1,
                                  const float* b1, const float* grn_g,
                                  const float* grn_b, const float* w2,
                                  const float* b2, hipStream_t stream) {
  dwconv7_kernel<<<dim3(TDIM / 256, CC, BB), 256, 0, stream>>>(x, dw_w, dw_b,
                                                               tmp, TDIM);
  layernorm_c_kernel<TDIM><<<dim3(TDIM / 512, BB), 128, 0, stream>>>(tmp, ln_g,
                                                                     ln_b);
  gemm_w1_gelu_kernel<TDIM><<<dim3(TDIM / 16, IC / 16, BB), 32, 0, stream>>>(
      tmp, w1, b1, H);
  rowsumsq_kernel<<<dim3(IC, BB), 256, 0, stream>>>(H, gx2, TDIM);
  grn_scale_kernel<<<BB, IC, 0, stream>>>(gx2, grn_g, scl);
  gemm_w2_grn_res_kernel<TDIM><<<dim3(TDIM / 16, CC / 16, BB), 32, 0, stream>>>(
      H, w2, b2, scl, grn_b, x, tmp);
  // block output lives in `tmp`
}

extern "C" void kernel_launch(void* const* d_in, const int* in_sizes, int n_in,
                              void* d_out, int out_size, void* d_ws,
                              size_t ws_size, hipStream_t stream) {
  (void)in_sizes; (void)n_in; (void)out_size; (void)ws_size;
  const float* queries = (const float*)d_in[0];
  const float* keys    = (const float*)d_in[1];
  const float* prior   = (const float*)d_in[2];
  // Param leaf order assumption (insertion order of the reference dicts):
  // key_params:   3=in_w 4=in_b ; blocks at 5+10*i:
  //   +0 dw_w +1 dw_b +2 ln_g +3 ln_b +4 w1 +5 b1 +6 grn_g +7 grn_b +8 w2 +9 b2
  // query_params: 35=in_w 36=in_b ; blocks at 37+10*i (same offsets)
  auto P = [&](int i) { return (const float*)d_in[i]; };

  float* ws = (float*)d_ws;
  float* Q0  = ws; ws += (size_t)BB * CC * T1C;  // 4.19M floats
  float* Q1  = ws; ws += (size_t)BB * CC * T1C;
  float* K0  = ws; ws += (size_t)BB * CC * T2C;  // 1.05M
  float* K1  = ws; ws += (size_t)BB * CC * T2C;
  float* H   = ws; ws += (size_t)BB * IC * T1C;  // 12.58M
  float* gx2 = ws; ws += (size_t)BB * IC;
  float* scl = ws; ws += (size_t)BB * IC;
  float* q2  = ws; ws += (size_t)BB * T1C;
  float* k2  = ws; ws += (size_t)BB * T2C;       // total ~92.5 MB

  // ---- query projection path ----
  proj_in_gemm<80, T1C><<<dim3(T1C / 16, CC / 16, BB), 32, 0, stream>>>(
      queries, P(35), P(36), Q0);
  float* qcur = Q0; float* qalt = Q1;
  for (int blk = 0; blk < 3; ++blk) {
    const int base = 37 + 10 * blk;
    launch_convnext_block<T1C>(qcur, qalt, H, gx2, scl, P(base + 0),
                               P(base + 1), P(base + 2), P(base + 3),
                               P(base + 4), P(base + 5), P(base + 6),
                               P(base + 7), P(base + 8), P(base + 9), stream);
    float* t = qcur; qcur = qalt; qalt = t;
  }

  // ---- key projection path ----
  proj_in_gemm<512, T2C><<<dim3(T2C / 16, CC / 16, BB), 32, 0, stream>>>(
      keys, P(3), P(4), K0);
  float* kcur = K0; float* kalt = K1;
  for (int blk = 0; blk < 3; ++blk) {
    const int base = 5 + 10 * blk;
    launch_convnext_block<T2C>(kcur, kalt, H, gx2, scl, P(base + 0),
                               P(base + 1), P(base + 2), P(base + 3),
                               P(base + 4), P(base + 5), P(base + 6),
                               P(base + 7), P(base + 8), P(base + 9), stream);
    float* t = kcur; kcur = kalt; kalt = t;
  }

  // ---- fused distance attention ----
  colsumsq_kernel<T1C><<<dim3(T1C / 512, BB), 128, 0, stream>>>(qcur, q2);
  colsumsq_kernel<T2C><<<dim3(T2C / 512, BB), 128, 0, stream>>>(kcur, k2);

  float* out_soft = (float*)d_out;
  float* out_logp = out_soft + (size_t)BB * T1C * T2C;
  attn_fused_kernel<<<dim3(T1C / 16, BB), 512, 0, stream>>>(
      qcur, kcur, q2, k2, prior, out_soft, out_logp);
}